// expert_layer_48473000903654
// MI455X (gfx1250) — compile-verified
//
#include <hip/hip_runtime.h>
#include <math.h>

// ---------------- problem constants ----------------
#define EE 4
#define BB 16
#define SS 256
#define CC 17
#define NN 16
#define DD 32
#define DFFN 64
#define HH 4
#define DHH 8

typedef float v2f __attribute__((ext_vector_type(2)));
typedef float v8f __attribute__((ext_vector_type(8)));
typedef unsigned int u32x4 __attribute__((ext_vector_type(4)));
typedef int i32x8 __attribute__((ext_vector_type(8)));
typedef int i32x4 __attribute__((ext_vector_type(4)));

// ---------------- WMMA f32 16x16x4 helpers ----------------
__device__ __forceinline__ v8f wmma4(v2f a, v2f b, v8f c) {
  // D = A(16x4,f32) * B(4x16,f32) + C(16x16,f32), full fp32 matrix pipe
  return __builtin_amdgcn_wmma_f32_16x16x4_f32(false, a, false, b, (short)0, c,
                                               false, false);
}

// C += A(16xK row-major, lda) * B(Kx16 row-major, ldb)
__device__ __forceinline__ v8f mm_AB(const float* A, int lda, const float* B,
                                     int ldb, int K, int lane) {
  v8f acc = {};
  int lr = lane & 15;
  int kh = (lane >> 4) << 1;  // upper half-wave handles K+2,K+3 of each quad
#pragma unroll
  for (int k0 = kh; k0 < K; k0 += 4) {
    v2f a, b;
    a.x = A[lr * lda + k0];
    a.y = A[lr * lda + k0 + 1];
    b.x = B[k0 * ldb + lr];
    b.y = B[(k0 + 1) * ldb + lr];
    acc = wmma4(a, b, acc);
  }
  return acc;
}

// C += A(16xK row-major, lda) * Bt(16xK row-major, ldbt)^T   (for Q*K^T)
__device__ __forceinline__ v8f mm_ABt(const float* A, int lda, const float* Bt,
                                      int ldbt, int K, int lane) {
  v8f acc = {};
  int lr = lane & 15;
  int kh = (lane >> 4) << 1;
#pragma unroll
  for (int k0 = kh; k0 < K; k0 += 4) {
    v2f a, b;
    a.x = A[lr * lda + k0];
    a.y = A[lr * lda + k0 + 1];
    b.x = Bt[lr * ldbt + k0];
    b.y = Bt[lr * ldbt + k0 + 1];
    acc = wmma4(a, b, acc);
  }
  return acc;
}

// ---------------- TDM: 1-D global->LDS DMA of nelem f32 ----------------
// D# group0: count=1, type=2(image), lds byte addr, 57-bit global addr.
// D# group1: data_size=4B, tensor_dim0=tile_dim0=nelem, tensor_dim1=tile_dim1=1.
// Groups 2/3 + trailing group zero (<=2D tensor). Tracked with TENSORcnt.
#if __has_builtin(__builtin_amdgcn_tensor_load_to_lds)
#define HAVE_TDM 1
__device__ __forceinline__ void tdm_load_1d(unsigned lds_byte,
                                            const float* gptr, int nelem) {
  unsigned long long ga = (unsigned long long)(uintptr_t)gptr;
  u32x4 g0;
  g0[0] = 1u;                     // count=1, is_restore=0, gather off
  g0[1] = lds_byte;               // lds_addr
  g0[2] = (unsigned)ga;           // global_addr[31:0]
  g0[3] = ((unsigned)(ga >> 32) & 0x01FFFFFFu) | 0x80000000u;  // addr hi | type=2
  i32x8 g1;
  g1[0] = 0x00020000;                               // data_size=4B
  g1[1] = (int)((nelem & 0xFFFF) << 16);            // tensor_dim0 lo16
  g1[2] = (int)(((nelem >> 16) & 0xFFFF) | (1 << 16));  // dim0 hi | tensor_dim1=1
  g1[3] = (int)((nelem & 0xFFFF) << 16);            // tile_dim0 = nelem
  g1[4] = 1;                                        // tile_dim1=1, tile_dim2=0
  g1[5] = nelem;                                    // tensor_dim0_stride lo32
  g1[6] = 0;
  g1[7] = 0;
  i32x4 z4 = {};
  i32x8 z8 = {};
  __builtin_amdgcn_tensor_load_to_lds(g0, g1, z4, z4, z8, 0);
}
#else
#define HAVE_TDM 0
#endif

// ---------------- Kernel A: series decomposition ----------------
// one block per (b, c) with c < N; 256 threads, thread s = sequence index
__global__ void decomp_kernel(const float* __restrict__ x,
                              float* __restrict__ new_x) {
  __shared__ float xs[SS];
  __shared__ float tc[SS], ts[SS];  // twiddles: cos/sin(2*pi*m/256)
  __shared__ float ar[129], ai[129], amp[129];
  __shared__ float thr;
  const int b = blockIdx.x >> 4;
  const int c = blockIdx.x & 15;
  const int s = threadIdx.x;

  xs[s] = x[(b * SS + s) * CC + c];
  {
    const float wp = 6.283185307179586f / 256.f;
    float sv, cv;
    sincosf(wp * (float)s, &sv, &cv);
    tc[s] = cv;
    ts[s] = sv;
  }
  __syncthreads();

  // rfft via table DFT: X[f] = sum_t x[t] * e^{-2pi i f t/256}
  if (s < 129) {
    float xr = 0.f, xi = 0.f;
    for (int t = 0; t < SS; ++t) {
      int m = (s * t) & 255;
      xr += xs[t] * tc[m];
      xi -= xs[t] * ts[m];
    }
    ar[s] = xr;
    ai[s] = xi;
    amp[s] = sqrtf(xr * xr + xi * xi);
  }
  __syncthreads();

  // 4th-largest amplitude over f=1..128 (DC excluded like amp[0]=-inf)
  if (s == 0) {
    float t0 = -3.4e38f, t1 = -3.4e38f, t2 = -3.4e38f, t3 = -3.4e38f;
    for (int f = 1; f < 129; ++f) {
      float a = amp[f];
      if (a > t0)      { t3 = t2; t2 = t1; t1 = t0; t0 = a; }
      else if (a > t1) { t3 = t2; t2 = t1; t1 = a; }
      else if (a > t2) { t3 = t2; t2 = a; }
      else if (a > t3) { t3 = a; }
    }
    thr = t3;
  }
  __syncthreads();

  // masked irfft: x[s] = (1/n) sum_f w_f * Re(X[f] e^{+2pi i f s/256})
  float acc = 0.f;
  for (int f = 1; f < 129; ++f) {
    if (amp[f] >= thr) {
      float wgt = (f == 128) ? 1.f : 2.f;
      int m = (f * s) & 255;
      acc += wgt * (ar[f] * tc[m] - ai[f] * ts[m]);
    }
  }
  float season = acc * (1.f / 256.f);

  // trend: mean of moving averages k = 4, 8, 12 with edge clamping
  float tr = 0.f;
  const int ks[3] = {4, 8, 12};
#pragma unroll
  for (int ki = 0; ki < 3; ++ki) {
    int k = ks[ki];
    int s0 = s - ((k - 1) >> 1);
    float a = 0.f;
    for (int j = 0; j < k; ++j) {
      int p = s0 + j;
      p = p < 0 ? 0 : (p > SS - 1 ? SS - 1 : p);
      a += xs[p];
    }
    tr += a / (float)k;
  }
  tr *= (1.f / 3.f);

  new_x[(b * SS + s) * NN + c] = xs[s] + season + tr;
}

// ---------------- Kernel B: expert assignment ----------------
__global__ void assign_kernel(const float* __restrict__ ox,
                              int* __restrict__ assign) {
  __shared__ float rmn[256], rmx[256];
  __shared__ float bins[EE + 1];
  const int t = threadIdx.x;
  float mn = 3.4e38f, mx = -3.4e38f;
  for (int i = t; i < BB * SS; i += 256) {
    float v = ox[i * 2 + 1];
    mn = fminf(mn, v);
    mx = fmaxf(mx, v);
  }
  rmn[t] = mn;
  rmx[t] = mx;
  __syncthreads();
  for (int off = 128; off; off >>= 1) {
    if (t < off) {
      rmn[t] = fminf(rmn[t], rmn[t + off]);
      rmx[t] = fmaxf(rmx[t], rmx[t + off]);
    }
    __syncthreads();
  }
  if (t == 0) {
    float lo = rmn[0], hi = rmx[0];
    float st = (hi - lo) * (1.f / (float)EE);
    for (int i = 0; i <= EE; ++i) bins[i] = lo + st * (float)i;
    bins[EE] = hi;  // linspace endpoint exact
  }
  __syncthreads();
  for (int i = t; i < BB * SS; i += 256) {
    float v = ox[i * 2 + 1];
    int cnt = 0;
#pragma unroll
    for (int j = 0; j <= EE; ++j) cnt += (bins[j] <= v) ? 1 : 0;  // searchsorted 'right'
    int a = cnt - 1;
    a = a < 0 ? 0 : (a > EE - 1 ? EE - 1 : a);
    assign[i] = a;
  }
}

// ---------------- Kernel C: full expert transformer block ----------------
// one block per (e, b, n); 256 threads = 8 wave32 waves; everything in LDS
#define LD 33    // padded stride for 32-wide rows (64 banks)
#define LDF 65   // padded stride for 64-wide rows
#define SCD 257  // padded stride for 256-wide score rows

#define H_OFF 0              // h residual stream, 256x33
#define Z_OFF 8448           // z / attention-out (aliased), 256x33
#define Q_OFF 16896          // q, 256x33 (also aliased by FFN f: 256x65)
#define K_OFF 25344          // k, 256x33
#define V_OFF 33792          // v, 256x33
#define SC_OFF 42240         // score strip 16x257
#define WQ_OFF 46352         // Wq 32x32 (Wk/Wv/Wo follow, 1024 apart)
#define W1_OFF 50448         // W1 32x64
#define W2_OFF 52496         // W2 64x32
#define P_OFF 54544          // b1(64) b2(32) l1g l1b l2g l2b (32 each)
#define ST_OFF 54768         // PV per-wave partials 8x16x8
#define SMEM_FLOATS 55792
#define F_OFF Q_OFF

__global__ void __launch_bounds__(256) expert_moe_kernel(
    const float* __restrict__ new_x, const int* __restrict__ assign,
    const float* __restrict__ start_W, const float* __restrict__ start_b,
    const float* __restrict__ Wq, const float* __restrict__ Wk,
    const float* __restrict__ Wv, const float* __restrict__ Wo,
    const float* __restrict__ l1g, const float* __restrict__ l1b,
    const float* __restrict__ l2g, const float* __restrict__ l2b,
    const float* __restrict__ W1, const float* __restrict__ b1,
    const float* __restrict__ W2, const float* __restrict__ b2,
    float* __restrict__ out) {
  extern __shared__ float sm[];
  const int tid = threadIdx.x;
  const int lane = tid & 31;
  const int wv = tid >> 5;
  const int e = blockIdx.x / (BB * NN);
  const int rem = blockIdx.x - e * (BB * NN);
  const int b = rem / NN;
  const int n = rem - b * NN;

  // ---- P0: stage weights (TDM DMA) + init hidden (overlapped) ----
#if HAVE_TDM
  if (wv == 0) {
    unsigned base = (unsigned)__builtin_amdgcn_groupstaticsize();
    tdm_load_1d(base + WQ_OFF * 4u,          Wq + e * DD * DD,   DD * DD);
    tdm_load_1d(base + (WQ_OFF + 1024) * 4u, Wk + e * DD * DD,   DD * DD);
    tdm_load_1d(base + (WQ_OFF + 2048) * 4u, Wv + e * DD * DD,   DD * DD);
    tdm_load_1d(base + (WQ_OFF + 3072) * 4u, Wo + e * DD * DD,   DD * DD);
    tdm_load_1d(base + W1_OFF * 4u,          W1 + e * DD * DFFN, DD * DFFN);
    tdm_load_1d(base + W2_OFF * 4u,          W2 + e * DFFN * DD, DFFN * DD);
  }
#else
  for (int i = tid; i < DD * DD; i += 256) {
    sm[WQ_OFF + i]        = Wq[e * DD * DD + i];
    sm[WQ_OFF + 1024 + i] = Wk[e * DD * DD + i];
    sm[WQ_OFF + 2048 + i] = Wv[e * DD * DD + i];
    sm[WQ_OFF + 3072 + i] = Wo[e * DD * DD + i];
  }
  for (int i = tid; i < DD * DFFN; i += 256) {
    sm[W1_OFF + i] = W1[e * DD * DFFN + i];
    sm[W2_OFF + i] = W2[e * DFFN * DD + i];
  }
#endif
  if (tid < DFFN) sm[P_OFF + tid] = b1[e * DFFN + tid];
  if (tid < DD) {
    sm[P_OFF + 64 + tid]  = b2[e * DD + tid];
    sm[P_OFF + 96 + tid]  = l1g[e * DD + tid];
    sm[P_OFF + 128 + tid] = l1b[e * DD + tid];
    sm[P_OFF + 160 + tid] = l2g[e * DD + tid];
    sm[P_OFF + 192 + tid] = l2b[e * DD + tid];
  }
  for (int idx = tid; idx < SS * DD; idx += 256) {
    int s = idx >> 5, d = idx & 31;
    float t = (assign[b * SS + s] == e) ? new_x[(b * SS + s) * NN + n] : 0.f;
    sm[H_OFF + s * LD + d] = t * start_W[d] + start_b[d];
  }
#if HAVE_TDM
  if (wv == 0) __builtin_amdgcn_s_wait_tensorcnt(0);
#endif
  __syncthreads();

  // ---- P1: LN1 (one row per thread) ----
  {
    const float* hr = &sm[H_OFF + tid * LD];
    float m = 0.f;
    for (int d = 0; d < DD; ++d) m += hr[d];
    m *= (1.f / DD);
    float va = 0.f;
    for (int d = 0; d < DD; ++d) { float t = hr[d] - m; va += t * t; }
    va *= (1.f / DD);
    float inv = rsqrtf(va + 1e-5f);
    float* zr = &sm[Z_OFF + tid * LD];
    for (int d = 0; d < DD; ++d)
      zr[d] = (hr[d] - m) * inv * sm[P_OFF + 96 + d] + sm[P_OFF + 128 + d];
  }
  __syncthreads();

  // ---- P2: QKV projections, 96 (16x16) tiles over 8 waves ----
  for (int job = wv; job < 96; job += 8) {
    int mat = job >> 5;  // 0=q 1=k 2=v
    int r2 = job & 31;
    int tm = r2 >> 1, tn = r2 & 1;
    v8f d = mm_AB(&sm[Z_OFF + tm * 16 * LD], LD,
                  &sm[WQ_OFF + mat * 1024 + tn * 16], DD, DD, lane);
    float* C = &sm[Q_OFF + mat * 8448 + tm * 16 * LD + tn * 16];
    int col = lane & 15, rb = (lane >> 4) * 8;
#pragma unroll
    for (int i = 0; i < 8; ++i) C[(rb + i) * LD + col] = d[i];
  }
  __syncthreads();

  // ---- P3: attention, unit = (head, 16-row q-tile), whole-WG cooperative ----
  const float iscale = 0.35355339059327373f;  // 1/sqrt(DH)
  for (int unit = 0; unit < 64; ++unit) {
    int head = unit >> 4, qt = unit & 15;
    // scores strip: 16 k-tiles across waves (Q tile 16x8 times K^T)
    for (int jt = wv; jt < 16; jt += 8) {
      v8f d = mm_ABt(&sm[Q_OFF + qt * 16 * LD + head * DHH], LD,
                     &sm[K_OFF + jt * 16 * LD + head * DHH], LD, DHH, lane);
      int col = lane & 15, rb = (lane >> 4) * 8;
#pragma unroll
      for (int i = 0; i < 8; ++i)
        sm[SC_OFF + (rb + i) * SCD + jt * 16 + col] = d[i] * iscale;
    }
    __syncthreads();
    // exact softmax over each of 16 rows, 16 lanes per row
    {
      int r = tid >> 4, j = tid & 15;
      float* sr = &sm[SC_OFF + r * SCD];
      float m = -3.4e38f;
      for (int c = j; c < SS; c += 16) m = fmaxf(m, sr[c]);
#pragma unroll
      for (int off = 1; off < 16; off <<= 1) m = fmaxf(m, __shfl_xor(m, off, 16));
      float ssum = 0.f;
      for (int c = j; c < SS; c += 16) ssum += expf(sr[c] - m);
#pragma unroll
      for (int off = 1; off < 16; off <<= 1) ssum += __shfl_xor(ssum, off, 16);
      float inv = 1.f / ssum;
      for (int c = j; c < SS; c += 16) sr[c] = expf(sr[c] - m) * inv;
    }
    __syncthreads();
    // P(16x256) @ V(256x8): each wave takes a 32-wide K chunk, partials to LDS
    {
      v8f d = {};
      int lr = lane & 15;
      int kh = (lane >> 4) << 1;
      int vc = head * DHH + (lr & 7);  // clamp duplicate cols 8..15 (discarded)
      const float* A = &sm[SC_OFF + wv * 32];
#pragma unroll
      for (int k0 = kh; k0 < 32; k0 += 4) {
        v2f a, bb;
        a.x = A[lr * SCD + k0];
        a.y = A[lr * SCD + k0 + 1];
        int kg = wv * 32 + k0;
        bb.x = sm[V_OFF + kg * LD + vc];
        bb.y = sm[V_OFF + (kg + 1) * LD + vc];
        d = wmma4(a, bb, d);
      }
      int col = lane & 15, rb = (lane >> 4) * 8;
      if (col < 8) {
#pragma unroll
        for (int i = 0; i < 8; ++i)
          sm[ST_OFF + wv * 128 + (rb + i) * 8 + col] = d[i];
      }
    }
    __syncthreads();
    // deterministic reduce of 8 wave partials into attention-out buffer
    if (tid < 128) {
      int r = tid >> 3, c2 = tid & 7;
      float a = 0.f;
#pragma unroll
      for (int w = 0; w < 8; ++w) a += sm[ST_OFF + w * 128 + r * 8 + c2];
      sm[Z_OFF + (qt * 16 + r) * LD + head * DHH + c2] = a;
    }
    __syncthreads();
  }

  // ---- P4: h += o @ Wo ----
  for (int job = wv; job < 32; job += 8) {
    int tm = job >> 1, tn = job & 1;
    v8f d = mm_AB(&sm[Z_OFF + tm * 16 * LD], LD,
                  &sm[WQ_OFF + 3072 + tn * 16], DD, DD, lane);
    float* C = &sm[H_OFF + tm * 16 * LD + tn * 16];
    int col = lane & 15, rb = (lane >> 4) * 8;
#pragma unroll
    for (int i = 0; i < 8; ++i) C[(rb + i) * LD + col] += d[i];
  }
  __syncthreads();

  // ---- P5: LN2 ----
  {
    const float* hr = &sm[H_OFF + tid * LD];
    float m = 0.f;
    for (int d = 0; d < DD; ++d) m += hr[d];
    m *= (1.f / DD);
    float va = 0.f;
    for (int d = 0; d < DD; ++d) { float t = hr[d] - m; va += t * t; }
    va *= (1.f / DD);
    float inv = rsqrtf(va + 1e-5f);
    float* zr = &sm[Z_OFF + tid * LD];
    for (int d = 0; d < DD; ++d)
      zr[d] = (hr[d] - m) * inv * sm[P_OFF + 160 + d] + sm[P_OFF + 192 + d];
  }
  __syncthreads();

  // ---- P6: f = relu(z @ W1 + b1)  (f overlays q/k region) ----
  for (int job = wv; job < 64; job += 8) {
    int tm = job >> 2, tn = job & 3;
    v8f d = mm_AB(&sm[Z_OFF + tm * 16 * LD], LD, &sm[W1_OFF + tn * 16], DFFN,
                  DD, lane);
    int col = lane & 15, rb = (lane >> 4) * 8;
#pragma unroll
    for (int i = 0; i < 8; ++i)
      sm[F_OFF + (tm * 16 + rb + i) * LDF + tn * 16 + col] =
          fmaxf(d[i] + sm[P_OFF + tn * 16 + col], 0.f);
  }
  __syncthreads();

  // ---- P7: h += f @ W2 + b2 ----
  for (int job = wv; job < 32; job += 8) {
    int tm = job >> 1, tn = job & 1;
    v8f d = mm_AB(&sm[F_OFF + tm * 16 * LDF], LDF, &sm[W2_OFF + tn * 16], DD,
                  DFFN, lane);
    float* C = &sm[H_OFF + tm * 16 * LD + tn * 16];
    int col = lane & 15, rb = (lane >> 4) * 8;
#pragma unroll
    for (int i = 0; i < 8; ++i)
      C[(rb + i) * LD + col] += d[i] + sm[P_OFF + 64 + tn * 16 + col];
  }
  __syncthreads();

  // ---- P8: scatter rows where this expert is assigned ----
  for (int idx = tid; idx < SS * DD; idx += 256) {
    int s = idx >> 5, d = idx & 31;
    if (assign[b * SS + s] == e)
      out[((b * SS + s) * NN + n) * DD + d] = sm[H_OFF + s * LD + d];
  }
}

// ---------------- host launcher ----------------
extern "C" void kernel_launch(void* const* d_in, const int* in_sizes, int n_in,
                              void* d_out, int out_size, void* d_ws,
                              size_t ws_size, hipStream_t stream) {
  (void)in_sizes; (void)n_in; (void)out_size; (void)ws_size;
  const float* x          = (const float*)d_in[0];
  const float* original_x = (const float*)d_in[1];
  const float* start_W    = (const float*)d_in[2];
  const float* start_b    = (const float*)d_in[3];
  const float* Wq         = (const float*)d_in[4];
  const float* Wk         = (const float*)d_in[5];
  const float* Wv         = (const float*)d_in[6];
  const float* Wo         = (const float*)d_in[7];
  const float* l1g        = (const float*)d_in[8];
  const float* l1b        = (const float*)d_in[9];
  const float* l2g        = (const float*)d_in[10];
  const float* l2b        = (const float*)d_in[11];
  const float* W1         = (const float*)d_in[12];
  const float* b1         = (const float*)d_in[13];
  const float* W2         = (const float*)d_in[14];
  const float* b2         = (const float*)d_in[15];
  float* out = (float*)d_out;

  float* nx  = (float*)d_ws;                                    // B*S*N f32
  int*   asg = (int*)((char*)d_ws + (size_t)BB * SS * NN * 4);  // B*S i32

  decomp_kernel<<<BB * NN, SS, 0, stream>>>(x, nx);
  assign_kernel<<<1, 256, 0, stream>>>(original_x, asg);
  expert_moe_kernel<<<EE * BB * NN, 256, SMEM_FLOATS * sizeof(float), stream>>>(
      nx, asg, start_W, start_b, Wq, Wk, Wv, Wo, l1g, l1b, l2g, l2b, W1, b1,
      W2, b2, out);
}